// CoOccurrenceLayer_49770081026133
// MI455X (gfx1250) — compile-verified
//
#include <hip/hip_runtime.h>
#include <hip/hip_bf16.h>
#include <stdint.h>

#define N_ 8
#define C_ 32
#define H_ 128
#define W_ 128
#define Q_ 16
#define TOT (N_*C_*H_*W_)

// tiling: each block computes TC x TH x TW outputs with 256 threads (8 waves)
#define TC 4
#define TH 4
#define TW 64
#define WPT 4
#define TILE_C (TC+2)
#define TILE_H (TH+2)
#define TILE_W (TW+2)
#define TILE_ELEMS (TILE_C*TILE_H*TILE_W)   // 6*6*66 = 2376 floats = 9504 B

// quantization index: trunc == floor because v >= xmin for real data;
// padded zeros may go negative but are clamped (and multiplied by x=0 anyway)
__device__ __forceinline__ int qidx(float v, float xmin, float scale) {
    int t = (int)((v - xmin) * scale);
    t = t < 0 ? 0 : t;
    t = t > (Q_ - 1) ? (Q_ - 1) : t;
    return t;
}

// CDNA5 async global -> LDS copy (GVS mode: saddr base + 32-bit vgpr byte offset)
__device__ __forceinline__ void async_ld_b32(const float* base, int byte_off, unsigned lds_off) {
    asm volatile("global_load_async_to_lds_b32 %0, %1, %2"
                 :
                 : "v"(lds_off), "v"(byte_off), "s"(base)
                 : "memory");
}
__device__ __forceinline__ void wait_async0() {
    asm volatile("s_wait_asynccnt 0" ::: "memory");
}

// ---------------- stage 1: per-block min/max partials ----------------
__global__ __launch_bounds__(256) void reduce1(const float* __restrict__ x,
                                               float* __restrict__ partial, int tot) {
    __shared__ float smn[256], smx[256];
    float mn =  3.402823466e+38f;
    float mx = -3.402823466e+38f;
    for (int i = blockIdx.x * 256 + threadIdx.x; i < tot; i += 256 * 256) {
        float v = x[i];
        mn = fminf(mn, v);
        mx = fmaxf(mx, v);
    }
    smn[threadIdx.x] = mn; smx[threadIdx.x] = mx;
    __syncthreads();
    for (int s = 128; s > 0; s >>= 1) {
        if (threadIdx.x < s) {
            smn[threadIdx.x] = fminf(smn[threadIdx.x], smn[threadIdx.x + s]);
            smx[threadIdx.x] = fmaxf(smx[threadIdx.x], smx[threadIdx.x + s]);
        }
        __syncthreads();
    }
    if (threadIdx.x == 0) {
        partial[blockIdx.x]       = smn[0];
        partial[256 + blockIdx.x] = smx[0];
    }
}

// ---------------- stage 2: final scalars {xmin, 16/xmax} ----------------
__global__ __launch_bounds__(256) void reduce2(const float* __restrict__ partial,
                                               float* __restrict__ scal) {
    __shared__ float smn[256], smx[256];
    smn[threadIdx.x] = partial[threadIdx.x];
    smx[threadIdx.x] = partial[256 + threadIdx.x];
    __syncthreads();
    for (int s = 128; s > 0; s >>= 1) {
        if (threadIdx.x < s) {
            smn[threadIdx.x] = fminf(smn[threadIdx.x], smn[threadIdx.x + s]);
            smx[threadIdx.x] = fmaxf(smx[threadIdx.x], smx[threadIdx.x + s]);
        }
        __syncthreads();
    }
    if (threadIdx.x == 0) {
        scal[0] = smn[0];                 // xmin
        scal[1] = (float)Q_ / smx[0];     // scale = q / xmax  (single division)
    }
}

// ---------------- main gather-stencil kernel ----------------
__global__ __launch_bounds__(256) void co_occ_main(const float* __restrict__ x,
                                                   const float* __restrict__ co,
                                                   const float* __restrict__ filt,
                                                   const float* __restrict__ scal,
                                                   float* __restrict__ out) {
    __shared__ float tile[TILE_ELEMS];
    __shared__ float coS[Q_ * Q_];

    const int tid = threadIdx.x;
    const int b   = blockIdx.x;
    const int wt  = b & 1;          // 2 w-tiles
    const int ht  = (b >> 1) & 31;  // 32 h-tiles
    const int ct  = (b >> 6) & 7;   // 8 c-tiles
    const int n   = b >> 9;         // 8 images

    if (tid < Q_ * Q_) coS[tid] = co[tid];

    // ---- stage halo tile into LDS via async-to-LDS; zero-fill OOB ----
    const float* xbase = x + (size_t)n * (C_ * H_ * W_);
    const int c0 = ct * TC - 1, h0 = ht * TH - 1, w0 = wt * TW - 1;

    for (int e = tid; e < TILE_ELEMS; e += 256) {
        int lc = e / (TILE_H * TILE_W);
        int r  = e - lc * (TILE_H * TILE_W);
        int lh = r / TILE_W;
        int lw = r - lh * TILE_W;
        int gc = c0 + lc, gh = h0 + lh, gw = w0 + lw;
        bool in = ((unsigned)gc < (unsigned)C_) &
                  ((unsigned)gh < (unsigned)H_) &
                  ((unsigned)gw < (unsigned)W_);
        if (in) {
            int goff = (((gc * H_) + gh) * W_ + gw) * 4;           // byte offset, >= 0
            unsigned lds = (unsigned)(uintptr_t)(&tile[e]);        // low 32 bits = LDS addr
            async_ld_b32(xbase, goff, lds);
        } else {
            tile[e] = 0.0f;  // disjoint addresses from async writes -> no race
        }
    }
    wait_async0();
    __syncthreads();

    // ---- filter in registers (27 uniform scalars) ----
    float f[27];
#pragma unroll
    for (int i = 0; i < 27; ++i) f[i] = filt[i];

    const float xmin  = scal[0];
    const float scale = scal[1];

    const int twg = tid & 15;         // 16 groups of WPT=4 along w
    const int th  = (tid >> 4) & 3;
    const int tc  = tid >> 6;
    const int wb  = twg * WPT;        // tile-w index of first output is wb+1

    int   rowb[WPT];
    float acc[WPT];
#pragma unroll
    for (int i = 0; i < WPT; ++i) {
        float xc = tile[((tc + 1) * TILE_H + (th + 1)) * TILE_W + wb + 1 + i];
        rowb[i] = qidx(xc, xmin, scale) << 4;   // row base into co matrix
        acc[i]  = 0.0f;
    }

#pragma unroll
    for (int dc = 0; dc < 3; ++dc) {
#pragma unroll
        for (int dh = 0; dh < 3; ++dh) {
            const float* lrow = &tile[((tc + dc) * TILE_H + (th + dh)) * TILE_W + wb];
            float xv[WPT + 2];
            int   iv[WPT + 2];
#pragma unroll
            for (int j = 0; j < WPT + 2; ++j) {
                xv[j] = lrow[j];
                iv[j] = qidx(xv[j], xmin, scale);
            }
#pragma unroll
            for (int dw = 0; dw < 3; ++dw) {
                const float fv = f[(dc * 3 + dh) * 3 + dw];
#pragma unroll
                for (int i = 0; i < WPT; ++i) {
                    // out += f * co[idx_center, idx_neighbor] * x_neighbor
                    acc[i] = fmaf(fv * xv[i + dw], coS[rowb[i] + iv[i + dw]], acc[i]);
                }
            }
        }
    }

    const int gw = wt * TW + wb;
    const int gh = ht * TH + th;
    const int gc = ct * TC + tc;
    size_t ob = (((size_t)n * C_ + gc) * H_ + gh) * W_ + gw;
    float4 o4 = make_float4(acc[0], acc[1], acc[2], acc[3]);
    *reinterpret_cast<float4*>(out + ob) = o4;   // 16B-aligned -> global_store_b128
}

extern "C" void kernel_launch(void* const* d_in, const int* in_sizes, int n_in,
                              void* d_out, int out_size, void* d_ws, size_t ws_size,
                              hipStream_t stream) {
    const float* x    = (const float*)d_in[0];  // (8,32,128,128) f32
    const float* co   = (const float*)d_in[1];  // (16,16) f32
    const float* filt = (const float*)d_in[2];  // (3,3,3) f32
    float* outp = (float*)d_out;
    float* wsf  = (float*)d_ws;   // [0..255] min partials, [256..511] max partials, [512..513] scalars

    reduce1<<<256, 256, 0, stream>>>(x, wsf, TOT);
    reduce2<<<1, 256, 0, stream>>>(wsf, wsf + 512);

    const int nblocks = (W_ / TW) * (H_ / TH) * (C_ / TC) * N_;  // 2*32*8*8 = 4096
    co_occ_main<<<nblocks, 256, 0, stream>>>(x, co, filt, wsf + 512, outp);
}